// LGATDirected_67336497266902
// MI455X (gfx1250) — compile-verified
//
#include <hip/hip_runtime.h>
#include <hip/hip_bf16.h>

// LGATDirected for MI455X (gfx1250, wave32).
// Phase 1: 4 attention hops (2 dirs x 2 hops), scatter/gather with HW atomics.
// Phase 2: fused concat-GEMM [N,768]x[768,64] using v_wmma_f32_16x16x4_f32.
//          W is transposed in LDS so every WMMA fragment is a single ds_load_b64.

#define N_NODES 100000
#define N_EDGES 600000
#define FDIM    128
#define ODIM    64

typedef float v2f __attribute__((ext_vector_type(2)));
typedef float v8f __attribute__((ext_vector_type(8)));

// ---- order-preserving float <-> u32 key (for exact segment-max via atomicMax) ----
__device__ __forceinline__ unsigned fkey(float f) {
    unsigned u = __float_as_uint(f);
    return (u & 0x80000000u) ? ~u : (u | 0x80000000u);
}
__device__ __forceinline__ float fdecode(unsigned k) {
    return (k & 0x80000000u) ? __uint_as_float(k ^ 0x80000000u) : __uint_as_float(~k);
}

// ---------------------------------------------------------------- zero scratch
__global__ __launch_bounds__(256) void zero_f(float* __restrict__ p, long n) {
    long i = (long)blockIdx.x * blockDim.x + threadIdx.x;
    if (i < n) p[i] = 0.0f;
}

// ------------------------------------------------- per-node dots: s = h.aS, d = h.aD
// one wave per node, lane covers 4 features (float4), shuffle reduction
__global__ __launch_bounds__(256) void node_scores(
    const float* __restrict__ h, const float* __restrict__ aS, const float* __restrict__ aD,
    float* __restrict__ sOut, float* __restrict__ dOut, int n)
{
    int node = (blockIdx.x * blockDim.x + threadIdx.x) >> 5;
    int lane = threadIdx.x & 31;
    if (node >= n) return;
    float4 hv  = *(const float4*)(h + (size_t)node * FDIM + lane * 4);
    float4 as4 = *(const float4*)(aS + lane * 4);
    float4 ad4 = *(const float4*)(aD + lane * 4);
    float ps = hv.x*as4.x + hv.y*as4.y + hv.z*as4.z + hv.w*as4.w;
    float pd = hv.x*ad4.x + hv.y*ad4.y + hv.z*ad4.z + hv.w*ad4.w;
    #pragma unroll
    for (int off = 16; off >= 1; off >>= 1) {
        ps += __shfl_xor(ps, off, 32);
        pd += __shfl_xor(pd, off, 32);
    }
    if (lane == 0) { sOut[node] = ps; dOut[node] = pd; }
}

// --------------------------- edge phase 1: e = leakyrelu(s[src]+d[dst]); segment max
__global__ __launch_bounds__(256) void edge_phase1(
    const int* __restrict__ src, const int* __restrict__ dst,
    const float* __restrict__ s, const float* __restrict__ d,
    float* __restrict__ e, unsigned* __restrict__ mkey, int nE)
{
    int i = blockIdx.x * blockDim.x + threadIdx.x;
    if (i >= nE) return;
    int di = dst[i];
    float x = s[src[i]] + d[di];
    float v = x > 0.0f ? x : 0.2f * x;
    e[i] = v;
    atomicMax(&mkey[di], fkey(v));
}

// ------------------------------- edge phase 2: ee = exp(e - m[dst]); segment sum
__global__ __launch_bounds__(256) void edge_phase2(
    const int* __restrict__ dst, float* __restrict__ e,
    const unsigned* __restrict__ mkey, float* __restrict__ denom, int nE)
{
    int i = blockIdx.x * blockDim.x + threadIdx.x;
    if (i >= nE) return;
    int di = dst[i];
    float ee = __expf(e[i] - fdecode(mkey[di]));   // m finite for any dst with edges
    e[i] = ee;
    atomicAdd(&denom[di], ee);
}

// ------------- edge phase 3: hnew[dst] += alpha * h[src]; wave per edge, lane = 4 feats
__global__ __launch_bounds__(256) void edge_aggregate(
    const int* __restrict__ src, const int* __restrict__ dst,
    const float* __restrict__ ee, const float* __restrict__ denom,
    const float* __restrict__ hprev, float* __restrict__ hnew, int nE)
{
    int edge = (blockIdx.x * blockDim.x + threadIdx.x) >> 5;
    int lane = threadIdx.x & 31;
    if (edge >= nE) return;
    int sI = src[edge], dI = dst[edge];
    float alpha = ee[edge] / (denom[dI] + 1e-16f);
    float4 hv = *(const float4*)(hprev + (size_t)sI * FDIM + lane * 4);
    float* base = hnew + (size_t)dI * FDIM + lane * 4;
    atomicAdd(base + 0, alpha * hv.x);
    atomicAdd(base + 1, alpha * hv.y);
    atomicAdd(base + 2, alpha * hv.z);
    atomicAdd(base + 3, alpha * hv.w);
}

// -------------------------------------------------------------------- final GEMM
// C[N,64] = concat(seg0..seg5)[N,768] @ W[768,64] + b, WMMA f32 16x16x4.
// Block = 256 thr (8 waves) -> 128 rows x 64 cols of C. k staged in 32-chunks via LDS.
// A kept row-major (stride 36), W stored TRANSPOSED (Wt[n][k], stride 34) so both the
// A fragment {A[m][k],A[m][k+1]} and B fragment {W[k][n],W[k+1][n]} are contiguous
// 8-byte LDS loads -> single ds_load_b64 each, no VGPR-pair assembly movs.
__global__ __launch_bounds__(256) void gemm_out(
    const float* __restrict__ s0, const float* __restrict__ s1, const float* __restrict__ s2,
    const float* __restrict__ s3, const float* __restrict__ s4, const float* __restrict__ s5,
    const float* __restrict__ W, const float* __restrict__ bias,
    float* __restrict__ out, int n)
{
    __shared__ __align__(16) float As[128 * 36];  // pad 36: conflict-free column reads
    __shared__ __align__(16) float Wt[64 * 34];   // transposed, pad 34 (rows 8B-aligned)
    const float* segs[6] = { s0, s1, s2, s3, s4, s5 };

    const int t     = threadIdx.x;
    const int wv    = t >> 5;
    const int lane  = t & 31;
    const int m     = lane & 15;        // A row-in-tile / B-N / C-N
    const int khalf = lane >> 4;        // K pair select
    const int rows0 = blockIdx.x * 128;

    v8f acc[4];
    #pragma unroll
    for (int nt = 0; nt < 4; ++nt)
        #pragma unroll
        for (int r = 0; r < 8; ++r) acc[nt][r] = 0.0f;

    #pragma unroll
    for (int seg = 0; seg < 6; ++seg) {
        const float* Aseg = segs[seg];
        for (int kc = 0; kc < FDIM; kc += 32) {
            // stage A: 128 rows x 32 cols, coalesced float4 (rows clamped, never OOB)
            #pragma unroll
            for (int i = 0; i < 4; ++i) {
                int idx4 = t + i * 256;          // 1024 float4
                int row  = idx4 >> 3;
                int c4   = (idx4 & 7) * 4;
                int rg   = rows0 + row; if (rg > n - 1) rg = n - 1;
                float4 v = *(const float4*)(Aseg + (size_t)rg * FDIM + kc + c4);
                *(float4*)(&As[row * 36 + c4]) = v;
            }
            // stage W transposed: global read coalesced float4, LDS scatter stores
            int kbase = seg * FDIM + kc;
            #pragma unroll
            for (int i = 0; i < 2; ++i) {
                int idx4 = t + i * 256;          // 512 float4 of W[32][64]
                int row  = idx4 >> 4;            // k within chunk
                int c4   = (idx4 & 15) * 4;      // n
                float4 v = *(const float4*)(W + (size_t)(kbase + row) * ODIM + c4);
                Wt[(c4 + 0) * 34 + row] = v.x;
                Wt[(c4 + 1) * 34 + row] = v.y;
                Wt[(c4 + 2) * 34 + row] = v.z;
                Wt[(c4 + 3) * 34 + row] = v.w;
            }
            __syncthreads();

            #pragma unroll
            for (int kk = 0; kk < 32; kk += 4) {
                int krow = kk + 2 * khalf;
                v2f a  = *(const v2f*)&As[(wv * 16 + m) * 36 + krow];
                v2f b0 = *(const v2f*)&Wt[(0 * 16 + m) * 34 + krow];
                v2f b1 = *(const v2f*)&Wt[(1 * 16 + m) * 34 + krow];
                v2f b2 = *(const v2f*)&Wt[(2 * 16 + m) * 34 + krow];
                v2f b3 = *(const v2f*)&Wt[(3 * 16 + m) * 34 + krow];
                acc[0] = __builtin_amdgcn_wmma_f32_16x16x4_f32(false, a, false, b0, (short)0, acc[0], false, false);
                acc[1] = __builtin_amdgcn_wmma_f32_16x16x4_f32(false, a, false, b1, (short)0, acc[1], false, false);
                acc[2] = __builtin_amdgcn_wmma_f32_16x16x4_f32(false, a, false, b2, (short)0, acc[2], false, false);
                acc[3] = __builtin_amdgcn_wmma_f32_16x16x4_f32(false, a, false, b3, (short)0, acc[3], false, false);
            }
            __syncthreads();
        }
    }

    // C/D layout: lanes 0-15 -> rows r, lanes 16-31 -> rows r+8; col = m
    const int rowBase = rows0 + wv * 16 + khalf * 8;
    #pragma unroll
    for (int nt = 0; nt < 4; ++nt) {
        float bv = bias[nt * 16 + m];
        #pragma unroll
        for (int r = 0; r < 8; ++r) {
            int row = rowBase + r;
            if (row < n) out[(size_t)row * ODIM + nt * 16 + m] = acc[nt][r] + bv;
        }
    }
}

// --------------------------------------------------------------------- host side
static void run_hop(const float* hprev, float* hnew,
                    const int* srcP, const int* dstP,
                    const float* aS, const float* aD,
                    float* sBuf, float* dBuf, unsigned* mkey, float* denom, float* eeBuf,
                    hipStream_t stream)
{
    long nf = (long)N_NODES * FDIM;
    zero_f<<<(unsigned)((nf + 255) / 256), 256, 0, stream>>>(hnew, nf);
    zero_f<<<(2 * N_NODES + 255) / 256, 256, 0, stream>>>((float*)mkey, 2L * N_NODES); // mkey+denom
    node_scores<<<(N_NODES + 7) / 8, 256, 0, stream>>>(hprev, aS, aD, sBuf, dBuf, N_NODES);
    edge_phase1<<<(N_EDGES + 255) / 256, 256, 0, stream>>>(srcP, dstP, sBuf, dBuf, eeBuf, mkey, N_EDGES);
    edge_phase2<<<(N_EDGES + 255) / 256, 256, 0, stream>>>(dstP, eeBuf, mkey, denom, N_EDGES);
    edge_aggregate<<<(N_EDGES + 7) / 8, 256, 0, stream>>>(srcP, dstP, eeBuf, denom, hprev, hnew, N_EDGES);
}

extern "C" void kernel_launch(void* const* d_in, const int* in_sizes, int n_in,
                              void* d_out, int out_size, void* d_ws, size_t ws_size,
                              hipStream_t stream)
{
    const float* feature = (const float*)d_in[0];
    const int*   ei      = (const int*)d_in[1];
    const float* att_f_s = (const float*)d_in[2];
    const float* att_f_d = (const float*)d_in[3];
    const float* att_r_s = (const float*)d_in[4];
    const float* att_r_d = (const float*)d_in[5];
    const float* W       = (const float*)d_in[6];
    const float* bias    = (const float*)d_in[7];
    float*       out     = (float*)d_out;

    const int* srcA = ei;            // edge_index[0]
    const int* dstA = ei + N_EDGES;  // edge_index[1]

    // workspace carve-up (~210 MB)
    size_t NF = (size_t)N_NODES * FDIM;
    float* h1f   = (float*)d_ws;
    float* h2f   = h1f + NF;
    float* h1r   = h2f + NF;
    float* h2r   = h1r + NF;
    float* sBuf  = h2r + NF;
    float* dBuf  = sBuf + N_NODES;
    unsigned* mkey = (unsigned*)(dBuf + N_NODES);
    float* denom = (float*)(mkey + N_NODES);
    float* eeBuf = denom + N_NODES;

    // forward direction: src -> dst
    run_hop(feature, h1f, srcA, dstA, att_f_s, att_f_d, sBuf, dBuf, mkey, denom, eeBuf, stream);
    run_hop(h1f,     h2f, srcA, dstA, att_f_s, att_f_d, sBuf, dBuf, mkey, denom, eeBuf, stream);
    // reverse direction: roles swapped (propagate(feature, dst, src, ...))
    run_hop(feature, h1r, dstA, srcA, att_r_s, att_r_d, sBuf, dBuf, mkey, denom, eeBuf, stream);
    run_hop(h1r,     h2r, dstA, srcA, att_r_s, att_r_d, sBuf, dBuf, mkey, denom, eeBuf, stream);

    // fused concat + GEMM + bias (concat order: [x, h1f, h2f, x, h1r, h2r])
    gemm_out<<<(N_NODES + 127) / 128, 256, 0, stream>>>(
        feature, h1f, h2f, feature, h1r, h2r, W, bias, out, N_NODES);
}